// EquivariantLayer_37520834298219
// MI455X (gfx1250) — compile-verified
//
#include <hip/hip_runtime.h>
#include <hip/hip_bf16.h>

// ---------------------------------------------------------------------------
// EquivariantLayer on MI455X (gfx1250): every spectral transform expressed as
// a batched f32 GEMM on V_WMMA_F32_16X16X4_F32, with the shared DFT-basis
// tile staged through LDS via gfx1250 async global->LDS loads.
// ---------------------------------------------------------------------------

typedef float v2f __attribute__((ext_vector_type(2)));
typedef float v8f __attribute__((ext_vector_type(8)));

#define TWO_PI 6.28318530717958647692f

// ---------------- sizes ----------------
#define NN1   128
#define NH    65
#define KYP   80          // padded ky (cols 65..79 of all bases are zero)
#define NN2   256
#define CC1   32
#define CC2   32
#define BB    8
#define CCIO  1024        // 32*32

// ---------------- workspace layout (float elements) ----------------
#define OFF_WFT_R 0L                    // [80][128]
#define OFF_WFT_I 10240L
#define OFF_CX_R  20480L                // [128][128]
#define OFF_CX_I  36864L
#define OFF_E_R   53248L                // [256][128]
#define OFF_E_I   86016L
#define OFF_V_R   118784L               // [256][80]
#define OFF_V_I   139264L
#define OFF_R3    163840L               // ksym [128][128][1024] then Fy/fhat/mix
#define OFF_FY_R  (OFF_R3)
#define OFF_FY_I  (OFF_R3 + 2621440L)
#define OFF_FH_R  (OFF_R3 + 5242880L)
#define OFF_FH_I  (OFF_R3 + 7864320L)
#define OFF_MIX_R (OFF_R3 + 10485760L)
#define OFF_MIX_I (OFF_R3 + 13107200L)
#define OFF_R1    (OFF_R3 + 16777216L)  // Ky [128][80][1024] r,i  -> later G
#define OFF_KY_R  (OFF_R1)
#define OFF_KY_I  (OFF_R1 + 10485760L)
#define OFF_G_R   (OFF_R1)
#define OFF_G_I   (OFF_R1 + 5242880L)
#define OFF_R2    (OFF_R1 + 20971520L)  // k_hat [128][80][1024] r,i
#define OFF_KH_R  (OFF_R2)
#define OFF_KH_I  (OFF_R2 + 10485760L)

// LDS tile geometry for the GEMM (A tiles: 16 rows x K<=128, row pad 4 floats)
#define LDS_ROWP 132
#define LDS_TILE (16 * LDS_ROWP)

// ---------------------------------------------------------------------------
// Basis tables: rDFT, full DFT, inverse row basis (IDX_ROW folded in),
// Hermitian-weighted irfft column basis.  Recomputed every launch (tiny).
// ---------------------------------------------------------------------------
__global__ void eq_basis_kernel(float* ws) {
  long i = (long)blockIdx.x * blockDim.x + threadIdx.x;
  if (i >= 159744L) return;
  if (i < 20480L) {                      // WfT: [k=80][y=128], e^{-2pi i y k /128}
    int isI = i >= 10240L; long j = isI ? i - 10240L : i;
    int k = (int)(j / 128), y = (int)(j % 128);
    float v = 0.f;
    if (k <= 64) {
      int t = (y * k) & 127;
      float ang = TWO_PI * (float)t / 128.f;
      v = isI ? -sinf(ang) : cosf(ang);
    }
    ws[(isI ? OFF_WFT_I : OFF_WFT_R) + j] = v;
  } else if (i < 53248L) {               // Cx: [kx=128][x=128], e^{-2pi i x kx /128}
    long j = i - 20480L; int isI = j >= 16384L; if (isI) j -= 16384L;
    int kx = (int)(j / 128), x = (int)(j % 128);
    int t = (kx * x) & 127;
    float ang = TWO_PI * (float)t / 128.f;
    ws[(isI ? OFF_CX_I : OFF_CX_R) + j] = isI ? -sinf(ang) : cosf(ang);
  } else if (i < 118784L) {              // E: [u=256][x=128], e^{+2pi i k(x) u /256}/256
    long j = i - 53248L; int isI = j >= 32768L; if (isI) j -= 32768L;
    int u = (int)(j / 128), x = (int)(j % 128);
    int m = (x <= 64) ? x : (x - 128);   // IDX_ROW frequency mapping
    int t = ((m * u) % 256 + 256) & 255;
    float ang = TWO_PI * (float)t / 256.f;
    float v = (isI ? sinf(ang) : cosf(ang)) * (1.f / 256.f);
    ws[(isI ? OFF_E_I : OFF_E_R) + j] = v;
  } else {                               // V: [n=256][k=80], hermitian irfft weights
    long j = i - 118784L; int isI = j >= 20480L; if (isI) j -= 20480L;
    int n = (int)(j / 80), k = (int)(j % 80);
    float v = 0.f;
    if (k <= 64) {
      float w = (k == 0) ? 1.f : 2.f;
      int t = (k * n) & 255;
      float ang = TWO_PI * (float)t / 256.f;
      v = (isI ? -sinf(ang) : cosf(ang)) * w * (1.f / 256.f);
    }
    ws[(isI ? OFF_V_I : OFF_V_R) + j] = v;
  }
}

// ---------------------------------------------------------------------------
// 4-fold rotational symmetrization of the kernel (float4 vectorized).
// k[x,y] = (K[x,y] + K[127-y,x] + K[127-x,127-y] + K[y,127-x]) / 4
// ---------------------------------------------------------------------------
__global__ void eq_ksym_kernel(const float4* __restrict__ K, float4* __restrict__ S) {
  long i = (long)blockIdx.x * blockDim.x + threadIdx.x;   // over float4 elems
  if (i >= 4194304L) return;
  int c4 = (int)(i & 255);
  long t = i >> 8;
  int y = (int)(t & 127);
  int x = (int)(t >> 7);
  float4 p0 = K[(long)(x * 128 + y) * 256 + c4];
  float4 p1 = K[(long)((127 - y) * 128 + x) * 256 + c4];
  float4 p2 = K[(long)((127 - x) * 128 + (127 - y)) * 256 + c4];
  float4 p3 = K[(long)(y * 128 + (127 - x)) * 256 + c4];
  float4 r;
  r.x = 0.25f * (p0.x + p1.x + p2.x + p3.x);
  r.y = 0.25f * (p0.y + p1.y + p2.y + p3.y);
  r.z = 0.25f * (p0.z + p1.z + p2.z + p3.z);
  r.w = 0.25f * (p0.w + p1.w + p2.w + p3.w);
  S[i] = r;
}

// ---------------------------------------------------------------------------
// Dual-product WMMA-f32 GEMM:  D = A1*B1 + s2*(A2*B2)   (batched, strided)
// - 8 waves/block along N, each wave computes NT consecutive 16x16 tiles
//   (one A fragment feeds NT WMMAs).
// - The block-shared A tiles (16 x K, both products) are staged into LDS with
//   gfx1250 async global->LDS B128 loads + s_wait_asynccnt, rows padded to
//   132 floats so the 16-lane fragment read hits distinct LDS banks.
// - B addressing strength-reduced to pointer bumps; inner loop is two-phase
//   (load all NT fragments into distinct locals, then issue NT WMMAs) so the
//   scheduler can clause the loads and stagger the waits in BOTH products.
// - All tile guards are wave-uniform: EXEC is all-ones inside WMMA.
// A frag: lane g=l>>4 holds K={2g,2g+1} of row (l&15).
// B frag: v.x/v.y = rows {2g,2g+1} at col (l&15).
// D frag: acc[r] = row r+8g, col (l&15).
// ---------------------------------------------------------------------------
template <int NT>
__global__ __launch_bounds__(256)
void eq_gemm2_kernel(const float* __restrict__ A1, const float* __restrict__ B1,
                     const float* __restrict__ A2, const float* __restrict__ B2,
                     float s2, float* __restrict__ D,
                     int M, int N, int K, int lda, int ldb, int ldd,
                     long bsA, long bsB, long bsD, int dual) {
  __shared__ float lds[2 * LDS_TILE];
  const int lane = threadIdx.x & 31;
  const int wave = threadIdx.x >> 5;
  const int mt = blockIdx.y;
  const long batch = blockIdx.z;

  const float* A1b = A1 + batch * bsA + (long)mt * 16 * lda;
  const float* A2b = dual ? (A2 + batch * bsA + (long)mt * 16 * lda) : A1b;

  // ---- async-stage A tiles (16 x K) into LDS (all threads participate) ----
  {
    const unsigned ldsbase = (unsigned)(unsigned long long)(const void*)lds;
    const int rowq = K >> 2;                 // 16-byte groups per row
    const int total = 16 * rowq;
    for (int e = threadIdx.x; e < total; e += 256) {
      int r = e / rowq, cq = e - r * rowq;
      unsigned l1 = ldsbase + (unsigned)((r * LDS_ROWP + cq * 4) * 4);
      const float* g1 = A1b + (long)r * lda + cq * 4;
      asm volatile("global_load_async_to_lds_b128 %0, %1, off"
                   :: "v"(l1), "v"(g1) : "memory");
      if (dual) {
        unsigned l2 = l1 + (unsigned)(LDS_TILE * 4);
        const float* g2 = A2b + (long)r * lda + cq * 4;
        asm volatile("global_load_async_to_lds_b128 %0, %1, off"
                     :: "v"(l2), "v"(g2) : "memory");
      }
    }
    asm volatile("s_wait_asynccnt 0x0" ::: "memory");
    __syncthreads();
  }

  const int nt0 = (blockIdx.x * 8 + wave) * NT;   // first N tile of this wave
  if ((nt0 + NT) * 16 > N) return;                // whole-wave exit (uniform)

  float* Db = D + batch * bsD;

  const int g  = lane >> 4;
  const int ml = lane & 15;
  const int col0 = nt0 * 16 + ml;
  const float* aRow1 = lds + ml * LDS_ROWP + (g << 1);   // 2g folded in
  const float* aRow2 = aRow1 + LDS_TILE;

  // per-lane B base pointers: column + 2g*ldb row offset folded in once
  const long ldbl   = (long)ldb;
  const long kstep  = 4 * ldbl;                    // pointer bump per k-step
  const long pfdist = 16 * ldbl;                   // prefetch distance
  const float* bA = B1 + batch * bsB + col0 + (ldbl * (g << 1));
  const float* bB = (dual ? B2 + batch * bsB : bA) + col0 * (dual ? 1 : 0);
  if (dual) bB += ldbl * (g << 1);

  v8f zero = {};
  v8f acc[NT];
#pragma unroll
  for (int t = 0; t < NT; ++t) acc[t] = zero;

#pragma unroll 2
  for (int kk = 0; kk < K; kk += 4) {
    if ((kk & 15) == 0 && kk + 16 < K)             // streamed-B prefetch
      __builtin_prefetch(bA + pfdist, 0, 1);

    // ---- phase 1: all fragment loads (distinct locals -> clauses) ----
    v2f a1 = *(const v2f*)(aRow1 + kk);            // ds_load_b64 (imm offset)
    v2f b1[NT];
#pragma unroll
    for (int t = 0; t < NT; ++t) {
      b1[t].x = bA[t * 16];
      b1[t].y = bA[t * 16 + ldbl];
    }
    v2f a2, b2[NT];
    if (dual) {
      a2 = *(const v2f*)(aRow2 + kk);
      a2.x *= s2; a2.y *= s2;                      // sign folded into A2
#pragma unroll
      for (int t = 0; t < NT; ++t) {
        b2[t].x = bB[t * 16];
        b2[t].y = bB[t * 16 + ldbl];
      }
      bB += kstep;
    }
    bA += kstep;

    // ---- phase 2: matrix ops ----
#pragma unroll
    for (int t = 0; t < NT; ++t)
      acc[t] = __builtin_amdgcn_wmma_f32_16x16x4_f32(false, a1, false, b1[t],
                                                     (short)0, acc[t], false, false);
    if (dual) {
#pragma unroll
      for (int t = 0; t < NT; ++t)
        acc[t] = __builtin_amdgcn_wmma_f32_16x16x4_f32(false, a2, false, b2[t],
                                                       (short)0, acc[t], false, false);
    }
  }

#pragma unroll
  for (int t = 0; t < NT; ++t) {
#pragma unroll
    for (int r = 0; r < 8; ++r) {
      const int orow = mt * 16 + r + 8 * g;
      Db[(long)orow * ldd + col0 + t * 16] = acc[t][r];
    }
  }
}

// ---------------------------------------------------------------------------
// Per-mode complex channel mix: mix[b,x,ky,o] = sum_i fh[b,x,ky,i]*kh[x,ky,i,o]
// minus real bias (real part only).  Padded ky>=65 written as zero.
// ---------------------------------------------------------------------------
__global__ void eq_mix_kernel(const float* __restrict__ fhr, const float* __restrict__ fhi,
                              const float* __restrict__ khr, const float* __restrict__ khi,
                              const float* __restrict__ bias,
                              float* __restrict__ mr, float* __restrict__ mi) {
  long i = (long)blockIdx.x * blockDim.x + threadIdx.x;
  if (i >= (long)BB * NN1 * KYP * CC2) return;
  int co = (int)(i & 31);
  long t = i >> 5;
  int ky = (int)(t % KYP); t /= KYP;
  int kx = (int)(t % NN1);
  int b  = (int)(t / NN1);
  long oidx = (((long)(b * NN1 + kx)) * KYP + ky) * 32 + co;
  if (ky >= NH) { mr[oidx] = 0.f; mi[oidx] = 0.f; return; }
  const float* fr = fhr + (((long)(b * NN1 + kx)) * KYP + ky) * 32;
  const float* fi = fhi + (((long)(b * NN1 + kx)) * KYP + ky) * 32;
  const float* kr = khr + (((long)kx * KYP + ky) * 1024) + co;
  const float* ki = khi + (((long)kx * KYP + ky) * 1024) + co;
  float sr = 0.f, si = 0.f;
#pragma unroll 4
  for (int ci = 0; ci < 32; ++ci) {
    float a = fr[ci], bq = fi[ci];
    float c = kr[ci * 32], d = ki[ci * 32];
    sr += a * c - bq * d;
    si += a * d + bq * c;
  }
  mr[oidx] = sr - bias[co];
  mi[oidx] = si;
}

// ---------------------------------------------------------------------------
static inline void launch_gemm2(hipStream_t s,
                                const float* A1, const float* B1,
                                const float* A2, const float* B2, float s2,
                                float* D, int M, int N, int K,
                                int lda, int ldb, int ldd,
                                long bsA, long bsB, long bsD,
                                int batches, int dual, int nt) {
  const int ntiles = N / 16;
  if (nt == 4) {
    dim3 grid((unsigned)((ntiles + 31) / 32), (unsigned)(M / 16), (unsigned)batches);
    eq_gemm2_kernel<4><<<grid, 256, 0, s>>>(A1, B1, A2, B2, s2, D,
                                            M, N, K, lda, ldb, ldd, bsA, bsB, bsD, dual);
  } else {
    dim3 grid((unsigned)((ntiles + 15) / 16), (unsigned)(M / 16), (unsigned)batches);
    eq_gemm2_kernel<2><<<grid, 256, 0, s>>>(A1, B1, A2, B2, s2, D,
                                            M, N, K, lda, ldb, ldd, bsA, bsB, bsD, dual);
  }
}

extern "C" void kernel_launch(void* const* d_in, const int* in_sizes, int n_in,
                              void* d_out, int out_size, void* d_ws, size_t ws_size,
                              hipStream_t stream) {
  const float* f    = (const float*)d_in[0];   // [8,128,128,32]
  const float* kin  = (const float*)d_in[1];   // [1,128,128,32,32]
  const float* bias = (const float*)d_in[2];   // [32]
  float* out = (float*)d_out;                  // [8,256,256,32]
  float* ws  = (float*)d_ws;

  // 0. DFT bases (tiny, recomputed each call: deterministic)
  eq_basis_kernel<<<(159744 + 255) / 256, 256, 0, stream>>>(ws);

  // 1. kernel symmetrization -> R3 (float4)
  eq_ksym_kernel<<<4194304 / 256, 256, 0, stream>>>((const float4*)kin,
                                                    (float4*)(ws + OFF_R3));

  // 2. kernel y-rDFT: Ky[x][ky][cc] = WfT * ksym[x]    (128 batches)
  launch_gemm2(stream, ws + OFF_WFT_R, ws + OFF_R3, nullptr, nullptr, 0.f,
               ws + OFF_KY_R, KYP, CCIO, NN1, 128, CCIO, CCIO,
               0L, 131072L, 81920L, 128, 0, 4);
  launch_gemm2(stream, ws + OFF_WFT_I, ws + OFF_R3, nullptr, nullptr, 0.f,
               ws + OFF_KY_I, KYP, CCIO, NN1, 128, CCIO, CCIO,
               0L, 131072L, 81920L, 128, 0, 4);

  // 3. kernel x-DFT (complex): k_hat = Cx * Ky    N = 80*1024
  launch_gemm2(stream, ws + OFF_CX_R, ws + OFF_KY_R, ws + OFF_CX_I, ws + OFF_KY_I, -1.f,
               ws + OFF_KH_R, NN1, KYP * CCIO, NN1, 128, KYP * CCIO, KYP * CCIO,
               0L, 0L, 0L, 1, 1, 4);
  launch_gemm2(stream, ws + OFF_CX_R, ws + OFF_KY_I, ws + OFF_CX_I, ws + OFF_KY_R, +1.f,
               ws + OFF_KH_I, NN1, KYP * CCIO, NN1, 128, KYP * CCIO, KYP * CCIO,
               0L, 0L, 0L, 1, 1, 4);

  // 4. f y-rDFT: Fy[(b,x)][ky][c] = WfT * f[b,x]   (1024 batches) -> R3 (ksym dead)
  launch_gemm2(stream, ws + OFF_WFT_R, f, nullptr, nullptr, 0.f,
               ws + OFF_FY_R, KYP, CC1, NN1, 128, 32, 32,
               0L, 4096L, 2560L, BB * NN1, 0, 2);
  launch_gemm2(stream, ws + OFF_WFT_I, f, nullptr, nullptr, 0.f,
               ws + OFF_FY_I, KYP, CC1, NN1, 128, 32, 32,
               0L, 4096L, 2560L, BB * NN1, 0, 2);

  // 5. f x-DFT (complex): f_hat[b] = Cx * Fy[b]    (8 batches, N = 80*32)
  launch_gemm2(stream, ws + OFF_CX_R, ws + OFF_FY_R, ws + OFF_CX_I, ws + OFF_FY_I, -1.f,
               ws + OFF_FH_R, NN1, KYP * CC1, NN1, 128, KYP * CC1, KYP * CC1,
               0L, 327680L, 327680L, BB, 1, 4);
  launch_gemm2(stream, ws + OFF_CX_R, ws + OFF_FY_I, ws + OFF_CX_I, ws + OFF_FY_R, +1.f,
               ws + OFF_FH_I, NN1, KYP * CC1, NN1, 128, KYP * CC1, KYP * CC1,
               0L, 327680L, 327680L, BB, 1, 4);

  // 6. per-mode complex channel mix minus bias (VALU; tiny FLOPs)
  {
    long total = (long)BB * NN1 * KYP * CC2;
    eq_mix_kernel<<<(unsigned)((total + 255) / 256), 256, 0, stream>>>(
        ws + OFF_FH_R, ws + OFF_FH_I, ws + OFF_KH_R, ws + OFF_KH_I, bias,
        ws + OFF_MIX_R, ws + OFF_MIX_I);
  }

  // 7. inverse row transform 128 -> 256 (complex): G[b] = E * mix[b]  (R1 reuse)
  launch_gemm2(stream, ws + OFF_E_R, ws + OFF_MIX_R, ws + OFF_E_I, ws + OFF_MIX_I, -1.f,
               ws + OFF_G_R, NN2, KYP * CC2, NN1, 128, KYP * CC2, KYP * CC2,
               0L, 327680L, 655360L, BB, 1, 4);
  launch_gemm2(stream, ws + OFF_E_R, ws + OFF_MIX_I, ws + OFF_E_I, ws + OFF_MIX_R, +1.f,
               ws + OFF_G_I, NN2, KYP * CC2, NN1, 128, KYP * CC2, KYP * CC2,
               0L, 327680L, 655360L, BB, 1, 4);

  // 8. column irfft: out[(b,u)][n][c] = Vr*Gr + Vi*Gi   (2048 batches, K=80)
  launch_gemm2(stream, ws + OFF_V_R, ws + OFF_G_R, ws + OFF_V_I, ws + OFF_G_I, +1.f,
               out, NN2, CC2, KYP, 80, 32, 32,
               0L, 2560L, 8192L, BB * NN2, 1, 2);

  (void)in_sizes; (void)n_in; (void)out_size; (void)ws_size;
}